// pa_lif_n_86260123173263
// MI455X (gfx1250) — compile-verified
//
#include <hip/hip_runtime.h>
#include <stdint.h>

#define T_STEPS 8
#define TAU     0.25f
#define THRESH  0.5f

#define THREADS 256
#define BLOCKS  512
#define NWAVES  (THREADS / 32)

// Per wave, per stage: 8 rows x 32 lanes x float4 = 4KB. Double buffered,
// 8 waves/block -> 64KB static LDS per block (<= 320KB/WGP).
__global__ __launch_bounds__(THREADS)
void pa_lif_async_kernel(const float* __restrict__ x, float* __restrict__ out,
                         int dim, int iters, int nthreads_total) {
    __shared__ float4 tile[2][NWAVES][T_STEPS][32];

    const int tid  = threadIdx.x;
    const int wv   = tid >> 5;
    const int lane = tid & 31;
    const int gid  = blockIdx.x * THREADS + tid;

    // Issue 8 async global->LDS b128 loads (one per timestep row) for tile `it`
    // into buffer `buf`. Per-wave private LDS region => no barrier needed,
    // only s_wait_asynccnt (in-order completion per wave).
    auto issue = [&](int buf, int it) {
        const size_t col = (size_t)(it * nthreads_total + gid) * 4;  // float index
#pragma unroll
        for (int t = 0; t < T_STEPS; ++t) {
            // Flat address of an LDS object: low 32 bits are the LDS byte offset.
            uint32_t loff = (uint32_t)(uintptr_t)(&tile[buf][wv][t][lane]);
            uint64_t ga   = (uint64_t)(uintptr_t)(x + (size_t)t * dim + col);
            asm volatile("global_load_async_to_lds_b128 %0, %1, off"
                         :: "v"(loff), "v"(ga)
                         : "memory");
        }
    };

    issue(0, 0);
    int buf = 0;

    for (int it = 0; it < iters; ++it) {
        if (it + 1 < iters) {
            issue(buf ^ 1, it + 1);
            // 16 outstanding max; first 8 (current buffer) done when cnt<=8.
            asm volatile("s_wait_asynccnt 0x8" ::: "memory");
        } else {
            asm volatile("s_wait_asynccnt 0x0" ::: "memory");
        }

        // Pull this wave's tile out of LDS (ds_load_b128 x 8).
        float xs[T_STEPS][4];
#pragma unroll
        for (int t = 0; t < T_STEPS; ++t) {
            float4 v = tile[buf][wv][t][lane];
            xs[t][0] = v.x; xs[t][1] = v.y; xs[t][2] = v.z; xs[t][3] = v.w;
        }

        float os[T_STEPS][4];
#pragma unroll
        for (int c = 0; c < 4; ++c) {
            // Scan 1: m[t] = sum_{j<=t} tau*(1-tau)^(t-j) x[j]  (== W_CONST @ x)
            // p1[t]  = sigmoid(0.5 - 0.5*(tau*x[t] + m[t])), clamped at 1e-12.
            float m = 0.0f;
            float P[T_STEPS];
#pragma unroll
            for (int t = 0; t < T_STEPS; ++t) {
                m = (1.0f - TAU) * m + TAU * xs[t][c];
                float z = THRESH - 0.5f * (TAU * xs[t][c] + m);
                float p = 1.0f / (1.0f + __expf(-z));
                P[t] = fmaxf(p, 1e-12f);
            }
            // Scan 2: MASK is the strict lower triangle, so
            //   pre[i] = tau*(x[i] + a[i]) - thresh,
            //   a[i]   = P[i-1]*tau*(a[i-1] + x[i-1]),  a[0] = 0.
            float a = 0.0f;
#pragma unroll
            for (int t = 0; t < T_STEPS; ++t) {
                float pre = TAU * (xs[t][c] + a) - THRESH;
                os[t][c] = (pre > 0.0f) ? 1.0f : 0.0f;
                a = P[t] * TAU * (a + xs[t][c]);
            }
        }

        const size_t col = (size_t)(it * nthreads_total + gid) * 4;
#pragma unroll
        for (int t = 0; t < T_STEPS; ++t) {
            float4 v = make_float4(os[t][0], os[t][1], os[t][2], os[t][3]);
            *(float4*)(out + (size_t)t * dim + col) = v;  // global_store_b128
        }
        buf ^= 1;
    }
}

extern "C" void kernel_launch(void* const* d_in, const int* in_sizes, int n_in,
                              void* d_out, int out_size, void* d_ws, size_t ws_size,
                              hipStream_t stream) {
    const float* x = (const float*)d_in[0];
    float* out = (float*)d_out;

    const int total = in_sizes[0];          // 8 * 128 * 16384
    const int dim   = total / T_STEPS;      // 2,097,152 (divisible by 4*BLOCKS*THREADS)
    const int nthreads = BLOCKS * THREADS;  // 131,072 lanes
    const int nv4   = dim / 4;              // 524,288 float4 columns
    const int iters = nv4 / nthreads;       // 4 tiles per thread

    pa_lif_async_kernel<<<BLOCKS, THREADS, 0, stream>>>(x, out, dim, iters, nthreads);
}